// MixHop_65326452572377
// MI455X (gfx1250) — compile-verified
//
#include <hip/hip_runtime.h>
#include <math.h>

#define N_NODES 50000
#define D_NODES 20000
#define M_NODES 30000
#define E_EDGES 800000
#define DSIM_ 512
#define MSIM_ 512
#define HID_ 128
#define OUT_ 128
#define PPOW 3
#define NPAIRS 100000

typedef float v2f __attribute__((ext_vector_type(2)));
typedef float v8f __attribute__((ext_vector_type(8)));

// ---------------- elementwise / graph kernels ----------------

__global__ void kzero4(float4* p, long long n4) {
    long long t = (long long)blockIdx.x * blockDim.x + threadIdx.x;
    if (t < n4) p[t] = make_float4(0.f, 0.f, 0.f, 0.f);
}

__global__ void kdeg(const int* __restrict__ edst, float* deg, int e) {
    int t = blockIdx.x * blockDim.x + threadIdx.x;
    if (t < e) atomicAdd(&deg[edst[t]], 1.0f);
}

__global__ void knorm(float* deg, int n) {
    int t = blockIdx.x * blockDim.x + threadIdx.x;
    if (t < n) {
        float d = deg[t];
        deg[t] = rsqrtf(d < 1.0f ? 1.0f : d);
    }
}

// out[row,:] = in[row,:] * nrm[row], vectorized float4 (wq = width/4)
__global__ void kscale(const float* __restrict__ in, float* __restrict__ out,
                       const float* __restrict__ nrm, int wq, long long n4) {
    long long t = (long long)blockIdx.x * blockDim.x + threadIdx.x;
    if (t >= n4) return;
    int row = (int)(t / wq);
    float s = nrm[row];
    float4 v = ((const float4*)in)[t];
    v.x *= s; v.y *= s; v.z *= s; v.w *= s;
    ((float4*)out)[t] = v;
}

// Y[dst[e],:] += X[src[e],:] * nrm[src[e]]  (float4 gather, source scale fused,
// 4x global_atomic_add_f32 into L2)
__global__ void kscatter(const int* __restrict__ esrc, const int* __restrict__ edst,
                         const float* __restrict__ x, const float* __restrict__ nrm,
                         float* y, int width, int wq, long long total) {
    long long t = (long long)blockIdx.x * blockDim.x + threadIdx.x;
    if (t >= total) return;
    int e = (int)(t / wq);
    int q = (int)(t % wq);
    int s = esrc[e], d = edst[e];
    float sc = nrm[s];
    float4 v = ((const float4*)(x + (long long)s * width))[q];
    float* yp = y + (long long)d * width + q * 4;
    atomicAdd(yp + 0, v.x * sc);
    atomicAdd(yp + 1, v.y * sc);
    atomicAdd(yp + 2, v.z * sc);
    atomicAdd(yp + 3, v.w * sc);
}

// ---------------- fp32 WMMA GEMM ----------------
// C[M,Ncols] = act( [A1(K1) | A2(K2)] @ W[Ncols, K1+K2]^T + bias )
// Optional per-row gather via idx1/idx2. One wave computes one 16x16 tile
// via V_WMMA_F32_16X16X4_F32 (A 16x4: lane<16 holds K0,K1; lane>=16 K2,K3).
// Two independent accumulator chains (even/odd K-steps) so consecutive WMMAs
// don't RAW-depend on each other's D operand.
// act: 0 = none, 1 = ELU, 2 = ReLU
__global__ __launch_bounds__(256) void kgemm(
    const float* __restrict__ A1, const int* __restrict__ idx1, int lda1, int K1,
    const float* __restrict__ A2, const int* __restrict__ idx2, int lda2, int K2,
    const float* __restrict__ W, int Ktot, const float* __restrict__ bias,
    float* __restrict__ C, int ldc, int M, int Ncols, int act)
{
    int wave = threadIdx.x >> 5;
    int lane = threadIdx.x & 31;
    int lid  = lane & 15;
    int half = lane >> 4;
    int rowBase = blockIdx.x << 4;
    int colBase = ((blockIdx.y << 3) + wave) << 4;
    if (colBase >= Ncols) return;     // uniform per-wave: EXEC stays all-ones

    int myRow = rowBase + lid;
    if (myRow >= M) myRow = M - 1;    // M is always a multiple of 16 here

    long long r1 = idx1 ? (long long)idx1[myRow] : (long long)myRow;
    const float* ap1 = A1 + r1 * (long long)lda1 + 2 * half;
    const float* wp  = W + (long long)(colBase + lid) * Ktot + 2 * half;

    v8f acc0 = {};
    v8f acc1 = {};
    {
        int k = 0;
        #pragma unroll 2
        for (; k + 8 <= K1; k += 8) {
            v2f a0 = *(const v2f*)(ap1 + k);
            v2f b0 = *(const v2f*)(wp + k);
            v2f a1 = *(const v2f*)(ap1 + k + 4);
            v2f b1 = *(const v2f*)(wp + k + 4);
            acc0 = __builtin_amdgcn_wmma_f32_16x16x4_f32(
                false, a0, false, b0, (short)0, acc0, false, false);
            acc1 = __builtin_amdgcn_wmma_f32_16x16x4_f32(
                false, a1, false, b1, (short)0, acc1, false, false);
        }
        for (; k < K1; k += 4) {
            v2f a = *(const v2f*)(ap1 + k);
            v2f b = *(const v2f*)(wp + k);
            acc0 = __builtin_amdgcn_wmma_f32_16x16x4_f32(
                false, a, false, b, (short)0, acc0, false, false);
        }
    }
    if (A2) {
        long long r2 = idx2 ? (long long)idx2[myRow] : (long long)myRow;
        const float* ap2 = A2 + r2 * (long long)lda2 + 2 * half;
        const float* wp2 = wp + K1;
        int k = 0;
        #pragma unroll 2
        for (; k + 8 <= K2; k += 8) {
            v2f a0 = *(const v2f*)(ap2 + k);
            v2f b0 = *(const v2f*)(wp2 + k);
            v2f a1 = *(const v2f*)(ap2 + k + 4);
            v2f b1 = *(const v2f*)(wp2 + k + 4);
            acc0 = __builtin_amdgcn_wmma_f32_16x16x4_f32(
                false, a0, false, b0, (short)0, acc0, false, false);
            acc1 = __builtin_amdgcn_wmma_f32_16x16x4_f32(
                false, a1, false, b1, (short)0, acc1, false, false);
        }
        for (; k < K2; k += 4) {
            v2f a = *(const v2f*)(ap2 + k);
            v2f b = *(const v2f*)(wp2 + k);
            acc0 = __builtin_amdgcn_wmma_f32_16x16x4_f32(
                false, a, false, b, (short)0, acc0, false, false);
        }
    }

    int col = colBase + lid;
    float bv = bias ? bias[col] : 0.0f;
    #pragma unroll
    for (int r = 0; r < 8; ++r) {
        int row = rowBase + (half << 3) + r;   // C/D layout: lanes 0-15 M=r, lanes 16-31 M=8+r
        if (row < M) {
            float v = acc0[r] + acc1[r] + bv;
            if (act == 1)      v = v > 0.f ? v : (expf(v) - 1.f);   // ELU
            else if (act == 2) v = fmaxf(v, 0.f);                    // ReLU
            C[(long long)row * ldc + col] = v;
        }
    }
}

// ---------------- predictor head: sigmoid(z1 . p1_w + p1_b) ----------------
__global__ void kpred(const float* __restrict__ z1, const float* __restrict__ w,
                      const float* __restrict__ b, float* __restrict__ out, int rows) {
    int row  = blockIdx.x * 8 + (threadIdx.x >> 5);
    int lane = threadIdx.x & 31;
    if (row >= rows) return;
    const float* z = z1 + (long long)row * HID_;
    float s = 0.f;
    #pragma unroll
    for (int j = 0; j < 4; ++j) {
        int c = lane + 32 * j;
        s += z[c] * w[c];
    }
    #pragma unroll
    for (int off = 16; off > 0; off >>= 1)
        s += __shfl_xor(s, off, 32);
    if (lane == 0) out[row] = 1.f / (1.f + expf(-(s + b[0])));
}

// ---------------- driver ----------------

extern "C" void kernel_launch(void* const* d_in, const int* in_sizes, int n_in,
                              void* d_out, int out_size, void* d_ws, size_t ws_size,
                              hipStream_t stream) {
    (void)in_sizes; (void)n_in; (void)out_size; (void)ws_size;
    const float* d_sim   = (const float*)d_in[0];
    const float* m_sim   = (const float*)d_in[1];
    const int*   esrc    = (const int*)d_in[2];
    const int*   edst    = (const int*)d_in[3];
    const int*   psrc    = (const int*)d_in[4];
    const int*   pdst    = (const int*)d_in[5];
    const float* d_fc_w  = (const float*)d_in[6];
    const float* d_fc_b  = (const float*)d_in[7];
    const float* m_fc_w  = (const float*)d_in[8];
    const float* m_fc_b  = (const float*)d_in[9];
    const float* l0_w    = (const float*)d_in[10];
    const float* l1_w    = (const float*)d_in[11];
    const float* fc_w    = (const float*)d_in[12];
    const float* d_fc1_w = (const float*)d_in[13];
    const float* d_fc1_b = (const float*)d_in[14];
    const float* m_fc1_w = (const float*)d_in[15];
    const float* m_fc1_b = (const float*)d_in[16];
    const float* p0_w    = (const float*)d_in[17];
    const float* p0_b    = (const float*)d_in[18];
    const float* p1_w    = (const float*)d_in[19];
    const float* p1_b    = (const float*)d_in[20];
    float* out = (float*)d_out;

    float* nrm = (float*)d_ws;                       // [N]
    float* B0  = nrm + N_NODES;                      // [N,384]
    float* B1  = B0 + (size_t)N_NODES * 384;         // [N,384]
    float* B2  = B1 + (size_t)N_NODES * 384;         // [N,384]

    dim3 blk(256);

    // 1) degrees -> norm = rsqrt(max(deg,1))
    {
        long long n4 = N_NODES / 4;
        kzero4<<<(unsigned)((n4 + 255) / 256), blk, 0, stream>>>((float4*)nrm, n4);
    }
    kdeg<<<(E_EDGES + 255) / 256, blk, 0, stream>>>(edst, nrm, E_EDGES);
    knorm<<<(N_NODES + 255) / 256, blk, 0, stream>>>(nrm, N_NODES);

    // 2) input projections -> B0 [N,128]
    kgemm<<<dim3(D_NODES / 16, 1), blk, 0, stream>>>(
        d_sim, nullptr, DSIM_, DSIM_, nullptr, nullptr, 0, 0,
        d_fc_w, DSIM_, d_fc_b, B0, HID_, D_NODES, HID_, 0);
    kgemm<<<dim3(M_NODES / 16, 1), blk, 0, stream>>>(
        m_sim + (size_t)D_NODES * MSIM_, nullptr, MSIM_, MSIM_, nullptr, nullptr, 0, 0,
        m_fc_w, MSIM_, m_fc_b, B0 + (size_t)D_NODES * HID_, HID_, M_NODES, HID_, 0);

    // 3) MixHop layers: for j in 0..2 emit X @ Wj^T into column block j of O;
    //    between hops: alt = scatter(X*norm[src] over edges); alt *= norm; swap.
    auto mixhop = [&](const float* Wstk, int win, float* X, float* Y, float* O) {
        int wq = win / 4;
        long long n4 = (long long)N_NODES * wq;
        unsigned gN = (unsigned)((n4 + 255) / 256);
        float* cur = X;
        float* alt = Y;
        for (int j = 0; j < PPOW; ++j) {
            kgemm<<<dim3(N_NODES / 16, 1), blk, 0, stream>>>(
                cur, nullptr, win, win, nullptr, nullptr, 0, 0,
                Wstk + (size_t)j * OUT_ * win, win, nullptr,
                O + j * OUT_, 384, N_NODES, OUT_, 0);
            if (j < PPOW - 1) {
                kzero4<<<gN, blk, 0, stream>>>((float4*)alt, n4);       // alt = 0
                long long tot = (long long)E_EDGES * wq;
                kscatter<<<(unsigned)((tot + 255) / 256), blk, 0, stream>>>(
                    esrc, edst, cur, nrm, alt, win, wq, tot);           // alt[dst] += cur[src]*nrm[src]
                kscale<<<gN, blk, 0, stream>>>(alt, alt, nrm, wq, n4);  // alt *= nrm
                float* t = cur; cur = alt; alt = t;
            }
        }
    };
    mixhop(l0_w, HID_, B0, B1, B2);   // B2 = layer0 concat [N,384]
    mixhop(l1_w, 384, B2, B1, B0);    // B0 = layer1 concat [N,384]

    // 4) fc: B1 = B0 @ fc_w^T  [N,128]
    kgemm<<<dim3(N_NODES / 16, 1), blk, 0, stream>>>(
        B0, nullptr, 384, 384, nullptr, nullptr, 0, 0,
        fc_w, 384, nullptr, B1, HID_, N_NODES, HID_, 0);

    // 5) fc1 with implicit concat(feats, sim), ELU -> B2 = h [N,128]
    kgemm<<<dim3(D_NODES / 16, 1), blk, 0, stream>>>(
        B1, nullptr, HID_, HID_, d_sim, nullptr, DSIM_, DSIM_,
        d_fc1_w, HID_ + DSIM_, d_fc1_b, B2, HID_, D_NODES, HID_, 1);
    kgemm<<<dim3(M_NODES / 16, 1), blk, 0, stream>>>(
        B1 + (size_t)D_NODES * HID_, nullptr, HID_, HID_,
        m_sim + (size_t)D_NODES * MSIM_, nullptr, MSIM_, MSIM_,
        m_fc1_w, HID_ + MSIM_, m_fc1_b, B2 + (size_t)D_NODES * HID_, HID_, M_NODES, HID_, 1);

    // 6) predictor layer0 with implicit concat(h[src], h[dst]), ReLU -> B0 [PAIRS,128]
    kgemm<<<dim3(NPAIRS / 16, 1), blk, 0, stream>>>(
        B2, psrc, HID_, HID_, B2, pdst, HID_, HID_,
        p0_w, 2 * HID_, p0_b, B0, HID_, NPAIRS, HID_, 2);

    // 7) predictor head: sigmoid(dot + b)
    kpred<<<(NPAIRS + 7) / 8, blk, 0, stream>>>(B0, p1_w, p1_b, out, NPAIRS);
}